// GCN_89498528514672
// MI455X (gfx1250) — compile-verified
//
#include <hip/hip_runtime.h>
#include <hip/hip_bf16.h>

// GCN on MI455X (gfx1250). Dense GEMMs use V_WMMA_F32_16X16X4_F32 (exact fp32
// matrix path, wave32). Sparse aggregation uses f32 global atomics; all
// scatter targets fit in the 192MB L2 so random traffic stays on-chip.

typedef float v2f __attribute__((ext_vector_type(2)));
typedef float v8f __attribute__((ext_vector_type(8)));

#define IN_DIM  128
#define HID_DIM 64
#define MID_DIM 128
#define OUT_DIM 32

// ---------------------------------------------------------------- elementwise
__global__ void zero_f32_kernel(float* __restrict__ p, int n) {
    int i = blockIdx.x * blockDim.x + threadIdx.x;
    if (i < n) p[i] = 0.0f;
}

// p[i] = bias[i % W]  (broadcast bias into accumulation buffer)
template <int W>
__global__ void bias_init_kernel(float* __restrict__ p, const float* __restrict__ bias, int n) {
    int i = blockIdx.x * blockDim.x + threadIdx.x;
    if (i < n) p[i] = bias[i % W];
}

__global__ void relu_kernel(float* __restrict__ p, int n) {
    int i = blockIdx.x * blockDim.x + threadIdx.x;
    if (i < n) p[i] = fmaxf(p[i], 0.0f);
}

// ----------------------------------------------------------------- degree
__global__ void deg_accum_kernel(const int* __restrict__ dst, float* __restrict__ deg, int E) {
    int i = blockIdx.x * blockDim.x + threadIdx.x;
    if (i < E) atomicAdd(&deg[dst[i]], 1.0f);
}

// dis = rsqrt(deg_edges + 1)   (+1 = self loop; deg >= 1 always so no zero case)
__global__ void rsqrt_kernel(float* __restrict__ deg, int n) {
    int i = blockIdx.x * blockDim.x + threadIdx.x;
    if (i < n) deg[i] = 1.0f / sqrtf(deg[i] + 1.0f);
}

// ----------------------------------------------------------------- WMMA GEMM
// C[n x OUTW] = A[n x K] @ B[K x OUTW] (+bias)(+relu)
// One wave computes a 16-row stripe across all OUTW columns (TILES accumulators).
// A: lane l holds M = l&15, K = k0 + 2*(l>>4) + v   (v2f)
// B: lane l holds N = l&15, K = k0 + 2*(l>>4) + v   (v2f)
// D: VGPR r -> row = r + 8*(l>>4), col = l&15       (v8f)
template <int K, int TILES, bool BIAS, bool RELU>
__global__ __launch_bounds__(256) void gemm_wmma_kernel(
    const float* __restrict__ A, const float* __restrict__ B,
    const float* __restrict__ bias, float* __restrict__ C,
    int numRowTiles, int n) {
    const int OUTW = TILES * 16;
    int wave = (blockIdx.x * blockDim.x + threadIdx.x) >> 5;
    int lane = threadIdx.x & 31;
    if (wave >= numRowTiles) return;              // wave-uniform: EXEC stays all-1s

    const int m    = lane & 15;
    const int half = lane >> 4;
    const int row0 = wave * 16;

    v8f acc[TILES] = {};

    int arow = row0 + m;
    if (arow >= n) arow = n - 1;                  // clamp (n is a multiple of 16 here)
    const float* __restrict__ ap = A + (long long)arow * K;

    for (int k0 = 0; k0 < K; k0 += 4) {
        v2f a = *(const v2f*)(ap + k0 + 2 * half);      // K = k0+2*half, k0+2*half+1
        const float* __restrict__ bp = B + (long long)(k0 + 2 * half) * OUTW + m;
#pragma unroll
        for (int t = 0; t < TILES; ++t) {
            v2f b;
            b.x = bp[t * 16];                            // K = k0+2*half
            b.y = bp[t * 16 + OUTW];                     // K = k0+2*half+1
            acc[t] = __builtin_amdgcn_wmma_f32_16x16x4_f32(
                false, a, false, b, (short)0, acc[t], false, false);
        }
    }

#pragma unroll
    for (int t = 0; t < TILES; ++t) {
        const int col = t * 16 + m;
        float bv = BIAS ? bias[col] : 0.0f;
#pragma unroll
        for (int r = 0; r < 8; ++r) {
            int row = row0 + r + 8 * half;
            if (row < n) {
                float v = acc[t][r] + bv;
                if (RELU) v = fmaxf(v, 0.0f);
                C[(long long)row * OUTW + col] = v;
            }
        }
    }
}

// ----------------------------------------------------------------- aggregation
// out[dst] += msg[src] * dis[src] * dis[dst]; edges e in [0,E) from arrays,
// e in [E, E+N) are self loops. One thread per (edge, 4-feature chunk).
template <int W>
__global__ void agg_kernel(const float* __restrict__ msg,
                           const int* __restrict__ src, const int* __restrict__ dst,
                           const float* __restrict__ dis, float* __restrict__ out,
                           int E, int N) {
    const int CH = W / 4;
    int idx = blockIdx.x * blockDim.x + threadIdx.x;
    int total = (E + N) * CH;
    if (idx >= total) return;
    int e = idx / CH;
    int c = (idx - e * CH) * 4;
    int s, d;
    if (e < E) { s = src[e]; d = dst[e]; }
    else       { s = d = e - E; }
    float nrm = dis[s] * dis[d];
    const float4 v = *(const float4*)(msg + (long long)s * W + c);
    float* o = out + (long long)d * W + c;
    atomicAdd(o + 0, v.x * nrm);
    atomicAdd(o + 1, v.y * nrm);
    atomicAdd(o + 2, v.z * nrm);
    atomicAdd(o + 3, v.w * nrm);
}

// ------------------------------------------------------------------- launcher
extern "C" void kernel_launch(void* const* d_in, const int* in_sizes, int n_in,
                              void* d_out, int out_size, void* d_ws, size_t ws_size,
                              hipStream_t stream) {
    const float* x   = (const float*)d_in[0];
    const int*   ei  = (const int*)  d_in[1];
    const float* W1  = (const float*)d_in[2];
    const float* b1  = (const float*)d_in[3];
    const float* Wl1 = (const float*)d_in[4];
    const float* bl1 = (const float*)d_in[5];
    const float* Wl2 = (const float*)d_in[6];
    const float* bl2 = (const float*)d_in[7];
    const float* W2  = (const float*)d_in[8];
    const float* b2  = (const float*)d_in[9];

    const int N = in_sizes[0] / IN_DIM;   // 100000
    const int E = in_sizes[1] / 2;        // 1600000
    const int* src = ei;
    const int* dst = ei + E;

    // workspace layout (floats): deg | bufA (N*128: xw1 then t) | bufB (N*64: h1 then h2w2) | bufC (N*64: h2)
    float* ws   = (float*)d_ws;
    float* deg  = ws;
    float* bufA = deg  + (size_t)N;
    float* bufB = bufA + (size_t)N * 128;
    float* bufC = bufB + (size_t)N * 64;
    float* out  = (float*)d_out;

    const int TPB = 256;
    const int rowTiles = (N + 15) / 16;
    const int gemmBlocks = (rowTiles + 7) / 8;     // 8 waves per 256-thread block

    // ---- symmetric normalization ---------------------------------------
    zero_f32_kernel<<<(N + TPB - 1) / TPB, TPB, 0, stream>>>(deg, N);
    deg_accum_kernel<<<(E + TPB - 1) / TPB, TPB, 0, stream>>>(dst, deg, E);
    rsqrt_kernel<<<(N + TPB - 1) / TPB, TPB, 0, stream>>>(deg, N);   // deg now holds dis

    // ---- conv1: xw1 = x @ W1 ; h1 = segsum + b1 ; relu -----------------
    gemm_wmma_kernel<IN_DIM, HID_DIM / 16, false, false>
        <<<gemmBlocks, TPB, 0, stream>>>(x, W1, nullptr, bufA, rowTiles, N);
    bias_init_kernel<HID_DIM><<<((N * HID_DIM) + TPB - 1) / TPB, TPB, 0, stream>>>(bufB, b1, N * HID_DIM);
    {
        int total = (E + N) * (HID_DIM / 4);
        agg_kernel<HID_DIM><<<(total + TPB - 1) / TPB, TPB, 0, stream>>>(bufA, src, dst, deg, bufB, E, N);
    }
    relu_kernel<<<((N * HID_DIM) + TPB - 1) / TPB, TPB, 0, stream>>>(bufB, N * HID_DIM);

    // ---- MLP: t = relu(h1@Wl1+bl1) ; h2 = relu(t@Wl2+bl2) --------------
    gemm_wmma_kernel<HID_DIM, MID_DIM / 16, true, true>
        <<<gemmBlocks, TPB, 0, stream>>>(bufB, Wl1, bl1, bufA, rowTiles, N);
    gemm_wmma_kernel<MID_DIM, HID_DIM / 16, true, true>
        <<<gemmBlocks, TPB, 0, stream>>>(bufA, Wl2, bl2, bufC, rowTiles, N);

    // ---- conv2: h2w2 = h2 @ W2 ; out = segsum + b2 ---------------------
    gemm_wmma_kernel<HID_DIM, OUT_DIM / 16, false, false>
        <<<gemmBlocks, TPB, 0, stream>>>(bufC, W2, nullptr, bufB, rowTiles, N);
    bias_init_kernel<OUT_DIM><<<((N * OUT_DIM) + TPB - 1) / TPB, TPB, 0, stream>>>(out, b2, N * OUT_DIM);
    {
        int total = (E + N) * (OUT_DIM / 4);
        agg_kernel<OUT_DIM><<<(total + TPB - 1) / TPB, TPB, 0, stream>>>(bufB, src, dst, deg, out, E, N);
    }
}